// PhysicsLayer_45535243272690
// MI455X (gfx1250) — compile-verified
//
#include <hip/hip_runtime.h>
#include <math.h>

#ifndef M_PI
#define M_PI 3.14159265358979323846
#endif

#define NXC 4096   // grid intervals; output has NXC+1 points
#define NTS 20     // Crank-Nicolson steps

typedef __attribute__((ext_vector_type(2))) float v2f;
typedef __attribute__((ext_vector_type(8))) float v8f;

// Z = X * Y for 16x16 f32 matrices living in LDS (row-major), where only the
// top-left 2x2 blocks of X and Y are nonzero -> a single K=4 WMMA is exact.
// Operand layouts per CDNA5 ISA 7.12.2:
//   A (16x4):  lane<16 -> row=lane, cols {0,1}; lane>=16 -> row=lane-16, cols {2,3}
//   B (4x16):  lane<16 -> col=lane, rows {0,1}; lane>=16 -> col=lane-16, rows {2,3}
//   D (16x16): vgpr v, lane -> row = v + 8*(lane>=16), col = lane&15
__device__ __forceinline__ void mm16(const float* __restrict__ X,
                                     const float* __restrict__ Y,
                                     float* __restrict__ Z, int lane) {
  const int half = lane >> 4;   // 0 or 1
  const int l    = lane & 15;
  v2f a, b;
  a.x = X[l * 16 + (2 * half + 0)];
  a.y = X[l * 16 + (2 * half + 1)];
  b.x = Y[(2 * half + 0) * 16 + l];
  b.y = Y[(2 * half + 1) * 16 + l];
  v8f c = {};
  // 8 args: (neg_a, A, neg_b, B, c_mod, C, reuse_a, reuse_b)
  v8f d = __builtin_amdgcn_wmma_f32_16x16x4_f32(false, a, false, b,
                                                (short)0, c, false, false);
  __syncthreads();
#pragma unroll
  for (int v = 0; v < 8; ++v) Z[(v + 8 * half) * 16 + l] = d[v];
  __syncthreads();
}

// One wave. Computes the 20-step mode-1 amplitude c20 via WMMA matrix
// squaring of the affine step operator M = [[g, s], [0, 1]].
__global__ void __launch_bounds__(32)
cn_evolve_operator(const float* __restrict__ pa, const float* __restrict__ pb,
                   const float* __restrict__ pt, float* __restrict__ ws) {
  __shared__ float buf0[256], buf1[256], buf2[256];
  const int lane = (int)threadIdx.x;

  // Spectral step coefficients (uniform across lanes; double to avoid the
  // 1-cos(pi/4096) f32 cancellation -- use 2*sin^2(theta/2)).
  const double alpha = (double)pa[0];
  const double beta  = (double)pb[0];
  const double tt    = (double)pt[0];
  const double dt  = tt / (double)NTS;
  const double dx  = 1.0 / (double)NXC;
  const double r   = alpha * dt / (dx * dx);
  const double th  = M_PI / (double)NXC;
  const double sh  = sin(0.5 * th);
  const double omc = 2.0 * sh * sh;          // 1 - cos(pi/NX), exact to f64
  const double lamA = 1.0 + r * omc;         // eigenvalue of A on mode 1
  const double lamB = 1.0 - r * omc;         // eigenvalue of B on mode 1
  const float g = (float)(lamB / lamA);
  const float s = (float)(dt * beta / lamA);

  for (int i = lane; i < 256; i += 32) buf0[i] = 0.0f;
  __syncthreads();
  if (lane == 0) { buf0[0] = g; buf0[1] = s; buf0[16 + 1] = 1.0f; }
  __syncthreads();

  mm16(buf0, buf0, buf1, lane);  // M^2
  mm16(buf1, buf1, buf2, lane);  // M^4  (kept in buf2)
  mm16(buf2, buf2, buf0, lane);  // M^8
  mm16(buf0, buf0, buf1, lane);  // M^16
  mm16(buf1, buf2, buf0, lane);  // M^20 = M^16 * M^4

  // [c20, 1]^T = M^20 [1, 1]^T  (c0 = 1)
  if (lane == 0) ws[0] = buf0[0] + buf0[1];
}

// Elementwise write: u[j] = c20 * sin(pi * j / NX), boundaries zero. 16 KB.
__global__ void __launch_bounds__(256)
cn_write_output(const float* __restrict__ ws, float* __restrict__ out) {
  const int j = (int)(blockIdx.x * blockDim.x + threadIdx.x);
  if (j > NXC) return;
  float v = 0.0f;
  if (j > 0 && j < NXC) {
    const float x = (float)j * (1.0f / (float)NXC);
    v = ws[0] * sinf((float)M_PI * x);
  }
  out[j] = v;
}

extern "C" void kernel_launch(void* const* d_in, const int* in_sizes, int n_in,
                              void* d_out, int out_size, void* d_ws, size_t ws_size,
                              hipStream_t stream) {
  (void)in_sizes; (void)n_in; (void)out_size; (void)ws_size;
  const float* pa = (const float*)d_in[0];  // alpha
  const float* pb = (const float*)d_in[1];  // beta
  const float* pt = (const float*)d_in[2];  // t
  float* ws  = (float*)d_ws;
  float* out = (float*)d_out;

  cn_evolve_operator<<<1, 32, 0, stream>>>(pa, pb, pt, ws);
  const int npts = NXC + 1;                 // 4097
  cn_write_output<<<(npts + 255) / 256, 256, 0, stream>>>(ws, out);
}